// TitansMemoryModule_56865366999258
// MI455X (gfx1250) — compile-verified
//
#include <hip/hip_runtime.h>
#include <math.h>

// ---------------------------------------------------------------------------
// Titans memory module for MI455X (gfx1250, wave32).
// Both large GEMMs run on the matrix pipe via V_WMMA_F32_16X16X4_F32 so that
// fp32 precision is preserved (inputs are fp32 normals; bf16 would be wrong).
// GEMM1: retrieved = k @ W^T           (32768x512 @ 512x512)
// GEMM2: grad      = diff^T @ k        (split-K over B=32768, f32 atomics)
// loss / column-means fused into GEMM2's loads; small elementwise kernels
// handle gates, momentum update and the two Frobenius-norm clips.
// ---------------------------------------------------------------------------

typedef __attribute__((ext_vector_type(2))) float v2f;
typedef __attribute__((ext_vector_type(8))) float v8f;

#define BATCH 32768
#define KD    512
#define VD    512

// workspace layout (floats)
#define WS_GRAD   0            // 512*512
#define WS_MCAND  262144       // 512*512
#define WS_WCAND  524288       // 512*512
#define WS_KSUM   786432       // 512
#define WS_VSUM   786944       // 512
#define WS_SCAL   787456       // [0]=loss_sum [1]=msq [2]=wsq [3]=alpha [4]=eta [5]=theta
#define WS_ZTAIL  1030         // floats to zero starting at WS_KSUM

// output layout (floats), concatenated in reference return order
#define OUT_RET   0
#define OUT_LOSS  16777216
#define OUT_NW    16777217
#define OUT_NM    17039361
#define OUT_GATES 17301505

#define TTT_LR 0.005f
#define MAXN   5.0f
#define EPS    1e-8f

// ---------------------------------------------------------------------------
// Zero accumulators (grad tile + sums + scalar reducers)
// ---------------------------------------------------------------------------
__global__ void k_zero(float* ws) {
    int i = blockIdx.x * blockDim.x + threadIdx.x;
    if (i < 262144)                ws[WS_GRAD + i] = 0.0f;
    else if (i < 262144 + WS_ZTAIL) ws[WS_KSUM + (i - 262144)] = 0.0f;
}

// ---------------------------------------------------------------------------
// GEMM 1: retrieved = k @ W^T.  One wave (32 thr) per 64x64 output tile.
// A frag (k, 16x4):  lane L: row = L&15, K = (L>>4)*2 + {0,1}  -> float2 load
// B frag (W^T, 4x16): lane L: col = L&15 (row of W), same K split -> float2
// ---------------------------------------------------------------------------
__global__ void __launch_bounds__(32)
k_gemm_retrieve(const float* __restrict__ k, const float* __restrict__ W,
                float* __restrict__ out) {
    const int lane = threadIdx.x;
    const int mr   = lane & 15;
    const int kh   = lane >> 4;            // 0/1
    const int m0   = blockIdx.x * 64;      // batch rows
    const int n0   = blockIdx.y * 64;      // val cols

    v8f acc[4][4] = {};
    for (int kk = 0; kk < KD; kk += 4) {
        v2f a[4], b[4];
#pragma unroll
        for (int i = 0; i < 4; ++i)
            a[i] = *(const v2f*)(k + (size_t)(m0 + 16 * i + mr) * KD + kk + kh * 2);
#pragma unroll
        for (int j = 0; j < 4; ++j)
            b[j] = *(const v2f*)(W + (size_t)(n0 + 16 * j + mr) * KD + kk + kh * 2);
#pragma unroll
        for (int i = 0; i < 4; ++i)
#pragma unroll
            for (int j = 0; j < 4; ++j)
                acc[i][j] = __builtin_amdgcn_wmma_f32_16x16x4_f32(
                    false, a[i], false, b[j], (short)0, acc[i][j], false, false);
    }
    // C/D layout: VGPR r, lane L -> row = (L>>4)*8 + r, col = L&15
#pragma unroll
    for (int i = 0; i < 4; ++i)
#pragma unroll
        for (int j = 0; j < 4; ++j)
#pragma unroll
            for (int r = 0; r < 8; ++r) {
                int m = m0 + 16 * i + kh * 8 + r;
                int n = n0 + 16 * j + mr;
                out[(size_t)m * VD + n] = acc[i][j][r];
            }
}

// ---------------------------------------------------------------------------
// GEMM 2 (split-K): grad = (2/(B*vd)) * diff^T @ k, diff = retrieved - v.
// grid (4,4,16), 128 threads = 2x2 waves, each wave does 64x64, K-chunk 2048.
// Fused: loss = sum(diff^2) (blocks y==0, wave-col 0) and column sums of
// v (same guard) and k (blocks x==0, wave-row 0) -> each element once.
// ---------------------------------------------------------------------------
__global__ void __launch_bounds__(128)
k_gemm_grad(const float* __restrict__ retr, const float* __restrict__ v,
            const float* __restrict__ k, float* __restrict__ ws) {
    const int lane = threadIdx.x & 31;
    const int wave = threadIdx.x >> 5;
    const int wm   = wave >> 1, wn = wave & 1;
    const int mr   = lane & 15;
    const int kh   = lane >> 4;

    const int m0 = blockIdx.x * 128 + wm * 64;   // val-dim rows of grad
    const int n0 = blockIdx.y * 128 + wn * 64;   // key-dim cols of grad
    const int b0 = blockIdx.z * 2048;            // batch chunk

    const bool doLoss = (blockIdx.y == 0) && (wn == 0);
    const bool doKsum = (blockIdx.x == 0) && (wm == 0);

    v8f   acc[4][4] = {};
    float lossAcc = 0.0f;
    float vpart[4] = {0.f, 0.f, 0.f, 0.f};
    float kpart[4] = {0.f, 0.f, 0.f, 0.f};

    for (int kk = 0; kk < 2048; kk += 4) {
        const int br0 = b0 + kk + kh * 2;        // this lane's two batch rows
        const int br1 = br0 + 1;
        v2f a[4], b[4];
#pragma unroll
        for (int i = 0; i < 4; ++i) {
            const int mi = m0 + 16 * i + mr;
            float r0 = retr[(size_t)br0 * VD + mi];
            float v0 = v   [(size_t)br0 * VD + mi];
            float r1 = retr[(size_t)br1 * VD + mi];
            float v1 = v   [(size_t)br1 * VD + mi];
            float d0 = r0 - v0, d1 = r1 - v1;
            a[i].x = d0; a[i].y = d1;
            if (doLoss) { lossAcc += d0 * d0 + d1 * d1; vpart[i] += v0 + v1; }
        }
#pragma unroll
        for (int j = 0; j < 4; ++j) {
            const int nj = n0 + 16 * j + mr;
            float k0 = k[(size_t)br0 * KD + nj];
            float k1 = k[(size_t)br1 * KD + nj];
            b[j].x = k0; b[j].y = k1;
            if (doKsum) kpart[j] += k0 + k1;
        }
#pragma unroll
        for (int i = 0; i < 4; ++i)
#pragma unroll
            for (int j = 0; j < 4; ++j)
                acc[i][j] = __builtin_amdgcn_wmma_f32_16x16x4_f32(
                    false, a[i], false, b[j], (short)0, acc[i][j], false, false);
    }

    if (doLoss) {
        atomicAdd(&ws[WS_SCAL + 0], lossAcc);
#pragma unroll
        for (int i = 0; i < 4; ++i)
            atomicAdd(&ws[WS_VSUM + m0 + 16 * i + mr], vpart[i]);
    }
    if (doKsum) {
#pragma unroll
        for (int j = 0; j < 4; ++j)
            atomicAdd(&ws[WS_KSUM + n0 + 16 * j + mr], kpart[j]);
    }

    const float gs = 2.0f / ((float)BATCH * (float)VD);
#pragma unroll
    for (int i = 0; i < 4; ++i)
#pragma unroll
        for (int j = 0; j < 4; ++j)
#pragma unroll
            for (int r = 0; r < 8; ++r) {
                int m = m0 + 16 * i + kh * 8 + r;
                int n = n0 + 16 * j + mr;
                atomicAdd(&ws[WS_GRAD + (size_t)m * KD + n], acc[i][j][r] * gs);
            }
}

// ---------------------------------------------------------------------------
// Gates: sigmoid([ksum/B ; vsum/B] @ gate_w^T + gate_b); also finalize loss.
// ---------------------------------------------------------------------------
__global__ void __launch_bounds__(256)
k_gates(const float* __restrict__ gw, const float* __restrict__ gb,
        float* __restrict__ ws, float* __restrict__ out) {
    __shared__ float red[3][256];
    const int t = threadIdx.x;
    float p0 = 0.f, p1 = 0.f, p2 = 0.f;
    const float invB = 1.0f / (float)BATCH;
    for (int j = t; j < KD + VD; j += 256) {
        float agg = (j < KD ? ws[WS_KSUM + j] : ws[WS_VSUM + (j - KD)]) * invB;
        p0 += agg * gw[0 * (KD + VD) + j];
        p1 += agg * gw[1 * (KD + VD) + j];
        p2 += agg * gw[2 * (KD + VD) + j];
    }
    red[0][t] = p0; red[1][t] = p1; red[2][t] = p2;
    __syncthreads();
    for (int s = 128; s > 0; s >>= 1) {
        if (t < s) {
            red[0][t] += red[0][t + s];
            red[1][t] += red[1][t + s];
            red[2][t] += red[2][t + s];
        }
        __syncthreads();
    }
    if (t == 0) {
        float alpha = 1.0f / (1.0f + expf(-(red[0][0] + gb[0])));
        float eta   = 1.0f / (1.0f + expf(-(red[1][0] + gb[1])));
        float theta = 1.0f / (1.0f + expf(-(red[2][0] + gb[2])));
        ws[WS_SCAL + 3] = alpha;
        ws[WS_SCAL + 4] = eta;
        ws[WS_SCAL + 5] = theta;
        out[OUT_GATES + 0] = alpha;
        out[OUT_GATES + 1] = eta;
        out[OUT_GATES + 2] = theta;
        out[OUT_LOSS] = ws[WS_SCAL + 0] / ((float)BATCH * (float)VD);
    }
}

// ---------------------------------------------------------------------------
// D1: m_cand = eta*S - lr*theta*clip(grad,-1,1); accumulate ||m_cand||^2
// ---------------------------------------------------------------------------
__global__ void __launch_bounds__(256)
k_mom1(const float* __restrict__ S, float* __restrict__ ws) {
    __shared__ float red[256];
    const int i = blockIdx.x * 256 + threadIdx.x;
    const float eta   = ws[WS_SCAL + 4];
    const float theta = ws[WS_SCAL + 5];
    float g = ws[WS_GRAD + i];
    g = fminf(1.0f, fmaxf(-1.0f, g));
    float m = eta * S[i] - TTT_LR * theta * g;
    ws[WS_MCAND + i] = m;
    red[threadIdx.x] = m * m;
    __syncthreads();
    for (int s = 128; s > 0; s >>= 1) {
        if (threadIdx.x < s) red[threadIdx.x] += red[threadIdx.x + s];
        __syncthreads();
    }
    if (threadIdx.x == 0) atomicAdd(&ws[WS_SCAL + 1], red[0]);
}

// ---------------------------------------------------------------------------
// D2: clip momentum norm, emit new_momentum; w_cand = (1-alpha)*W + nm;
//     accumulate ||w_cand||^2
// ---------------------------------------------------------------------------
__global__ void __launch_bounds__(256)
k_mom2(const float* __restrict__ W, float* __restrict__ ws,
       float* __restrict__ out) {
    __shared__ float red[256];
    const int i = blockIdx.x * 256 + threadIdx.x;
    const float alpha = ws[WS_SCAL + 3];
    const float nM = sqrtf(ws[WS_SCAL + 1]);
    const float sM = (nM > MAXN) ? (MAXN / (nM + EPS)) : 1.0f;
    float nm = sM * ws[WS_MCAND + i];
    out[OUT_NM + i] = nm;
    float w = (1.0f - alpha) * W[i] + nm;
    ws[WS_WCAND + i] = w;
    red[threadIdx.x] = w * w;
    __syncthreads();
    for (int s = 128; s > 0; s >>= 1) {
        if (threadIdx.x < s) red[threadIdx.x] += red[threadIdx.x + s];
        __syncthreads();
    }
    if (threadIdx.x == 0) atomicAdd(&ws[WS_SCAL + 2], red[0]);
}

// ---------------------------------------------------------------------------
// D3: clip weight norm, emit new_weight
// ---------------------------------------------------------------------------
__global__ void __launch_bounds__(256)
k_mom3(float* __restrict__ ws, float* __restrict__ out) {
    const int i = blockIdx.x * 256 + threadIdx.x;
    const float nW = sqrtf(ws[WS_SCAL + 2]);
    const float sW = (nW > MAXN) ? (MAXN / (nW + EPS)) : 1.0f;
    out[OUT_NW + i] = sW * ws[WS_WCAND + i];
}

// ---------------------------------------------------------------------------
extern "C" void kernel_launch(void* const* d_in, const int* in_sizes, int n_in,
                              void* d_out, int out_size, void* d_ws, size_t ws_size,
                              hipStream_t stream) {
    const float* k_in  = (const float*)d_in[0];   // [B, 512]
    const float* v_in  = (const float*)d_in[1];   // [B, 512]
    const float* mem_w = (const float*)d_in[2];   // [512, 512]
    const float* gw    = (const float*)d_in[3];   // [3, 1024]
    const float* gb    = (const float*)d_in[4];   // [3]
    const float* mom_S = (const float*)d_in[5];   // [512, 512]
    float* out = (float*)d_out;
    float* ws  = (float*)d_ws;

    // 0) zero accumulators
    k_zero<<<dim3((262144 + WS_ZTAIL + 255) / 256), dim3(256), 0, stream>>>(ws);

    // 1) retrieved = k @ W^T  (WMMA f32)
    k_gemm_retrieve<<<dim3(BATCH / 64, VD / 64), dim3(32), 0, stream>>>(
        k_in, mem_w, out + OUT_RET);

    // 2) grad (split-K WMMA f32) + fused loss / column sums
    k_gemm_grad<<<dim3(4, 4, 16), dim3(128), 0, stream>>>(
        out + OUT_RET, v_in, k_in, ws);

    // 3) gates + loss finalize
    k_gates<<<dim3(1), dim3(256), 0, stream>>>(gw, gb, ws, out);

    // 4-6) momentum / weight updates with Frobenius norm clips
    k_mom1<<<dim3(262144 / 256), dim3(256), 0, stream>>>(mom_S, ws);
    k_mom2<<<dim3(262144 / 256), dim3(256), 0, stream>>>(mem_w, ws, out);
    k_mom3<<<dim3(262144 / 256), dim3(256), 0, stream>>>(ws, out);
}